// SixConv_57157424775211
// MI455X (gfx1250) — compile-verified
//
#include <hip/hip_runtime.h>
#include <math.h>

// Problem constants (match reference file)
#define N_NODES 50000
#define N_EDGES 1600000
#define TOT_EDGES (N_EDGES + N_NODES)   // + self loops

typedef __attribute__((ext_vector_type(2))) float v2f;
typedef __attribute__((ext_vector_type(8))) float v8f;

// -------------------------------------------------------------------------
// h0 = relu(x)
__global__ void relu_copy_kernel(const float* __restrict__ x,
                                 float* __restrict__ h, int n) {
    int i = blockIdx.x * blockDim.x + threadIdx.x;
    if (i < n) h[i] = fmaxf(x[i], 0.f);
}

// -------------------------------------------------------------------------
// deg[d] = #in-edges (incl. self loop); graph shared by all 5 layers
__global__ void degree_kernel(const int* __restrict__ dst,
                              float* __restrict__ deg) {
    int e = blockIdx.x * blockDim.x + threadIdx.x;
    if (e >= TOT_EDGES) return;
    int d = (e < N_EDGES) ? dst[e] : (e - N_EDGES);
    atomicAdd(&deg[d], 1.0f);
}

// -------------------------------------------------------------------------
// Node GEMM with CDNA5 fp32 WMMA:   y[N, OUTF] = h[N, K] @ W[K, OUTF]
// plus (H>1)                        p[N, H]    = h[N, K] @ u[K, H]
// One wave32 per 16x16 output tile. EXEC is all-ones through every WMMA.
//
// V_WMMA_F32_16X16X4_F32 fragment layout (ISA 7.12.2):
//   A (16x4, MxK): lane = M (mod 16); lanes 0-15 hold K=k0,k0+1 in v[0],v[1];
//                  lanes 16-31 hold K=k0+2,k0+3.
//   C/D (16x16):   VGPR r: lanes 0-15 -> M=r, N=lane; lanes 16-31 -> M=r+8.
//   B (4x16, KxN): mirrored from A (lane = N, half selects K pair).
template<int K, int OUTF, int H>
__global__ __launch_bounds__(32)
void node_gemm_wmma(const float* __restrict__ h, const float* __restrict__ W,
                    const float* __restrict__ u, float* __restrict__ y,
                    float* __restrict__ p) {
    const int NT   = OUTF / 16;
    int tile    = blockIdx.x;
    int rowTile = tile / NT;
    int nt      = tile % NT;
    int lane    = threadIdx.x;
    int half    = lane >> 4;      // selects K pair within quad
    int l16     = lane & 15;
    int m0      = rowTile * 16;
    int n0      = nt * 16;

    const float* arow = h + (size_t)(m0 + l16) * K;

    v8f acc = {};
#pragma unroll
    for (int k0 = 0; k0 < K; k0 += 4) {
        int ka = k0 + half * 2;
        v2f a, b;
        a.x = arow[ka];
        a.y = arow[ka + 1];
        b.x = W[(size_t)ka * OUTF + n0 + l16];
        b.y = W[(size_t)(ka + 1) * OUTF + n0 + l16];
        acc = __builtin_amdgcn_wmma_f32_16x16x4_f32(
            /*neg_a=*/false, a, /*neg_b=*/false, b,
            /*c_mod=*/(short)0, acc, /*reuse_a=*/false, /*reuse_b=*/false);
    }

#pragma unroll
    for (int r = 0; r < 8; ++r) {
        int row = m0 + r + half * 8;
        y[(size_t)row * OUTF + n0 + l16] = acc[r];
    }

    // Attention logits p = h @ u (only meaningful when H > 1). Runs after all
    // WMMAs so divergence here cannot violate the EXEC-all-ones requirement.
    if (H > 1 && nt == 0 && half == 0) {
        int row = m0 + l16;
#pragma unroll
        for (int hh = 0; hh < H; ++hh) {
            float s = 0.f;
            for (int k = 0; k < K; ++k) s += arow[k] * u[k * H + hh];
            p[row * H + hh] = s;
        }
    }
}

// -------------------------------------------------------------------------
// H = 1 layers: softmax over one head == 1, so message is y[src] directly.
// One thread per (edge, channel); atomics resolve in L2 (s fits in 192MB L2).
template<int OUT>
__global__ void edge_scatter_h1(const int* __restrict__ src,
                                const int* __restrict__ dst,
                                const float* __restrict__ y,
                                float* __restrict__ sacc) {
    int t = blockIdx.x * blockDim.x + threadIdx.x;
    if (t >= TOT_EDGES * OUT) return;
    int e = t / OUT, ch = t % OUT;
    int s = (e < N_EDGES) ? src[e] : (e - N_EDGES);
    int d = (e < N_EDGES) ? dst[e] : (e - N_EDGES);
    atomicAdd(&sacc[(size_t)d * OUT + ch], y[(size_t)s * OUT + ch]);
}

// H = 4 layers: q = softmax(p[src] - p[dst] + c) over 4 heads,
// m[ch] = sum_h q_h * y[src][h*OUT + ch].
template<int OUT>
__global__ void edge_scatter_h4(const int* __restrict__ src,
                                const int* __restrict__ dst,
                                const float* __restrict__ p,
                                const float* __restrict__ c,
                                const float* __restrict__ y,
                                float* __restrict__ sacc) {
    int t = blockIdx.x * blockDim.x + threadIdx.x;
    if (t >= TOT_EDGES * OUT) return;
    int e = t / OUT, ch = t % OUT;
    int s = (e < N_EDGES) ? src[e] : (e - N_EDGES);
    int d = (e < N_EDGES) ? dst[e] : (e - N_EDGES);

    const float* ps = p + s * 4;
    const float* pd = p + d * 4;
    float a0 = ps[0] - pd[0] + c[0];
    float a1 = ps[1] - pd[1] + c[1];
    float a2 = ps[2] - pd[2] + c[2];
    float a3 = ps[3] - pd[3] + c[3];
    float mx = fmaxf(fmaxf(a0, a1), fmaxf(a2, a3));
    float e0 = __expf(a0 - mx), e1 = __expf(a1 - mx);
    float e2 = __expf(a2 - mx), e3 = __expf(a3 - mx);
    float inv = 1.f / (e0 + e1 + e2 + e3);

    const float* yr = y + (size_t)s * (4 * OUT) + ch;
    float m = (e0 * yr[0] + e1 * yr[OUT] + e2 * yr[2 * OUT] + e3 * yr[3 * OUT]) * inv;
    atomicAdd(&sacc[(size_t)d * OUT + ch], m);
}

// -------------------------------------------------------------------------
// h_next = relu(s / max(deg,1) + b)
template<int OUT>
__global__ void finalize_kernel(const float* __restrict__ sacc,
                                const float* __restrict__ deg,
                                const float* __restrict__ b,
                                float* __restrict__ hn) {
    int t = blockIdx.x * blockDim.x + threadIdx.x;
    if (t >= N_NODES * OUT) return;
    int i = t / OUT, ch = t % OUT;
    hn[t] = fmaxf(sacc[t] / fmaxf(deg[i], 1.f) + b[ch], 0.f);
}

// -------------------------------------------------------------------------
// MLP head: 64 -> 16 -> 4 -> 1 (+ sigmoid). Weights cached in LDS (4.4KB).
__global__ __launch_bounds__(256)
void mlp_kernel(const float* __restrict__ h,
                const float* __restrict__ lw1, const float* __restrict__ lb1,
                const float* __restrict__ lw2, const float* __restrict__ lb2,
                const float* __restrict__ lw3, const float* __restrict__ lb3,
                float* __restrict__ out) {
    __shared__ float w1[64 * 16];
    __shared__ float b1s[16], w2[16 * 4], b2s[4], w3s[4];
    __shared__ float b3s;
    for (int k = threadIdx.x; k < 1024; k += 256) w1[k] = lw1[k];
    if (threadIdx.x < 16) b1s[threadIdx.x] = lb1[threadIdx.x];
    if (threadIdx.x < 64) w2[threadIdx.x] = lw2[threadIdx.x];
    if (threadIdx.x < 4) { b2s[threadIdx.x] = lb2[threadIdx.x];
                           w3s[threadIdx.x] = lw3[threadIdx.x]; }
    if (threadIdx.x == 0) b3s = lb3[0];
    __syncthreads();

    int i = blockIdx.x * 256 + threadIdx.x;
    if (i >= N_NODES) return;

    float hv[64];
#pragma unroll
    for (int k = 0; k < 64; ++k) hv[k] = h[(size_t)i * 64 + k];

    float t1[16];
#pragma unroll
    for (int j = 0; j < 16; ++j) {
        float a = b1s[j];
        for (int k = 0; k < 64; ++k) a += hv[k] * w1[k * 16 + j];
        t1[j] = fmaxf(a, 0.f);
    }
    float t2[4];
#pragma unroll
    for (int j = 0; j < 4; ++j) {
        float a = b2s[j];
        for (int k = 0; k < 16; ++k) a += t1[k] * w2[k * 4 + j];
        t2[j] = fmaxf(a, 0.f);
    }
    float z = b3s;
#pragma unroll
    for (int k = 0; k < 4; ++k) z += t2[k] * w3s[k];
    out[i] = 1.f / (1.f + __expf(-z));
}

// -------------------------------------------------------------------------
extern "C" void kernel_launch(void* const* d_in, const int* in_sizes, int n_in,
                              void* d_out, int out_size, void* d_ws, size_t ws_size,
                              hipStream_t stream) {
    const float* x   = (const float*)d_in[0];
    const int*   ei  = (const int*)d_in[1];
    const int*   src = ei;
    const int*   dst = ei + N_EDGES;
    const float *W2 = (const float*)d_in[2],  *u2 = (const float*)d_in[3],
                *c2 = (const float*)d_in[4],  *b2 = (const float*)d_in[5];
    const float *W3 = (const float*)d_in[6],  *u3 = (const float*)d_in[7],
                *c3 = (const float*)d_in[8],  *b3 = (const float*)d_in[9];
    const float *W4 = (const float*)d_in[10], *u4 = (const float*)d_in[11],
                *b4 = (const float*)d_in[13];
    const float *W5 = (const float*)d_in[14], *u5 = (const float*)d_in[15],
                *b5 = (const float*)d_in[17];
    const float *W6 = (const float*)d_in[18], *u6 = (const float*)d_in[19],
                *b6 = (const float*)d_in[21];
    const float *lw1 = (const float*)d_in[22], *lb1 = (const float*)d_in[23];
    const float *lw2 = (const float*)d_in[24], *lb2 = (const float*)d_in[25];
    const float *lw3 = (const float*)d_in[26], *lb3 = (const float*)d_in[27];

    // Workspace layout (floats): 50000*(1+4+4*64) = 13.05M floats = 52.2 MB
    float* ws  = (float*)d_ws;
    size_t off = 0;
    float* deg = ws + off; off += (size_t)N_NODES;
    float* p   = ws + off; off += (size_t)N_NODES * 4;
    float* hA  = ws + off; off += (size_t)N_NODES * 64;
    float* hB  = ws + off; off += (size_t)N_NODES * 64;
    float* y   = ws + off; off += (size_t)N_NODES * 64;
    float* sac = ws + off; off += (size_t)N_NODES * 64;

    const int ROWT = N_NODES / 16;          // 3125 (exact)
    const int EB   = (TOT_EDGES + 255) / 256;

    // Degree (same graph for all layers)
    hipMemsetAsync(deg, 0, (size_t)N_NODES * sizeof(float), stream);
    degree_kernel<<<EB, 256, 0, stream>>>(dst, deg);

    // h0 = relu(x)
    relu_copy_kernel<<<(N_NODES * 16 + 255) / 256, 256, 0, stream>>>(x, hA, N_NODES * 16);

    // ---- conv2: 16 -> 16, H=4 (hA -> hB)
    node_gemm_wmma<16, 64, 4><<<ROWT * 4, 32, 0, stream>>>(hA, W2, u2, y, p);
    hipMemsetAsync(sac, 0, (size_t)N_NODES * 16 * sizeof(float), stream);
    edge_scatter_h4<16><<<(TOT_EDGES * 16 + 255) / 256, 256, 0, stream>>>(src, dst, p, c2, y, sac);
    finalize_kernel<16><<<(N_NODES * 16 + 255) / 256, 256, 0, stream>>>(sac, deg, b2, hB);

    // ---- conv3: 16 -> 16, H=4 (hB -> hA)
    node_gemm_wmma<16, 64, 4><<<ROWT * 4, 32, 0, stream>>>(hB, W3, u3, y, p);
    hipMemsetAsync(sac, 0, (size_t)N_NODES * 16 * sizeof(float), stream);
    edge_scatter_h4<16><<<(TOT_EDGES * 16 + 255) / 256, 256, 0, stream>>>(src, dst, p, c3, y, sac);
    finalize_kernel<16><<<(N_NODES * 16 + 255) / 256, 256, 0, stream>>>(sac, deg, b3, hA);

    // ---- conv4: 16 -> 16, H=1 (hA -> hB)   softmax(1 head) == 1
    node_gemm_wmma<16, 16, 1><<<ROWT * 1, 32, 0, stream>>>(hA, W4, u4, y, p);
    hipMemsetAsync(sac, 0, (size_t)N_NODES * 16 * sizeof(float), stream);
    edge_scatter_h1<16><<<(TOT_EDGES * 16 + 255) / 256, 256, 0, stream>>>(src, dst, y, sac);
    finalize_kernel<16><<<(N_NODES * 16 + 255) / 256, 256, 0, stream>>>(sac, deg, b4, hB);

    // ---- conv5: 16 -> 32, H=1 (hB -> hA)
    node_gemm_wmma<16, 32, 1><<<ROWT * 2, 32, 0, stream>>>(hB, W5, u5, y, p);
    hipMemsetAsync(sac, 0, (size_t)N_NODES * 32 * sizeof(float), stream);
    edge_scatter_h1<32><<<(TOT_EDGES * 32 + 255) / 256, 256, 0, stream>>>(src, dst, y, sac);
    finalize_kernel<32><<<(N_NODES * 32 + 255) / 256, 256, 0, stream>>>(sac, deg, b5, hA);

    // ---- conv6: 32 -> 64, H=1 (hA -> hB)
    node_gemm_wmma<32, 64, 1><<<ROWT * 4, 32, 0, stream>>>(hA, W6, u6, y, p);
    hipMemsetAsync(sac, 0, (size_t)N_NODES * 64 * sizeof(float), stream);
    edge_scatter_h1<64><<<(TOT_EDGES * 64 + 255) / 256, 256, 0, stream>>>(src, dst, y, sac);
    finalize_kernel<64><<<(N_NODES * 64 + 255) / 256, 256, 0, stream>>>(sac, deg, b6, hB);

    // ---- MLP head + sigmoid
    mlp_kernel<<<(N_NODES + 255) / 256, 256, 0, stream>>>(hB, lw1, lb1, lw2, lb2, lw3, lb3,
                                                          (float*)d_out);
}